// MS_SPS_79216376807580
// MI455X (gfx1250) — compile-verified
//
#include <hip/hip_runtime.h>
#include <hip/hip_bf16.h>
#include <stdint.h>

// ---- problem dims (from reference) ----
#define T_DIM 32
#define B_DIM 16
#define C_DIM 2
#define V_DIM 256
#define E_DIM 256
#define BN_EPS 1e-5f

// ---- tiling ----
#define ETILE 64     // e-chains per workgroup (== blockDim.x, 2 waves)
#define VT    64     // v-block staged per phase
#define VTP   65     // padded LDS row stride (bank-conflict-free column access)
#define XCHUNK 8     // t's per TDM chunk (8*512 floats = 16 KB LDS)

typedef float v2f  __attribute__((ext_vector_type(2)));
typedef float v8f  __attribute__((ext_vector_type(8)));
typedef unsigned int u32x4 __attribute__((ext_vector_type(4)));
typedef int   i32x8 __attribute__((ext_vector_type(8)));
typedef int   i32x4 __attribute__((ext_vector_type(4)));

__global__ __launch_bounds__(ETILE) void sps_lif_kernel(
    const float* __restrict__ x,     // [T,B,C,V]
    const float* __restrict__ W,     // [E,C]
    const float* __restrict__ gamma, // [E]
    const float* __restrict__ beta,  // [E]
    const float* __restrict__ rmean, // [E]
    const float* __restrict__ rvar,  // [E]
    float* __restrict__ out)         // [T,B,E,V]
{
  __shared__ float xbuf[XCHUNK * C_DIM * V_DIM]; // [t%8][c][v] via TDM
  __shared__ float pre [ETILE * VTP];            // preactivation tile [e_local][v]
  __shared__ float sbuf[ETILE * VTP];            // spike tile [e_local][v]

  const int tid  = threadIdx.x;
  const int lane = tid & 31;
  const int wv   = tid >> 5;               // wave id: 0 or 1
  const int e0   = blockIdx.x * ETILE;
  const int b    = blockIdx.y;

  // ---- fold BN into WMMA operands: A = scale*W, C = shift (broadcast over v) ----
  v2f aops[2];
  v8f cops[2];
  const int off = (lane >= 16) ? 8 : 0;    // C/D row offset for upper half-wave
  #pragma unroll
  for (int i = 0; i < 2; ++i) {
    const int te = wv * 2 + i;             // this wave's e-subtiles
    const int eb = e0 + te * 16;
    if (lane < 16) {                       // A lanes 0-15: K=0,1 ; lanes 16-31: K=2,3 (zero pad)
      const int e = eb + lane;
      const float sc = gamma[e] * rsqrtf(rvar[e] + BN_EPS);
      aops[i].x = W[e * C_DIM + 0] * sc;
      aops[i].y = W[e * C_DIM + 1] * sc;
    } else {
      aops[i].x = 0.f; aops[i].y = 0.f;
    }
    #pragma unroll
    for (int g = 0; g < 8; ++g) {
      const int e = eb + g + off;
      const float sc = gamma[e] * rsqrtf(rvar[e] + BN_EPS);
      cops[i][g] = beta[e] - rmean[e] * sc;
    }
  }

  float vstate = 0.f;                      // LIF membrane for chain e = e0 + tid

  for (int t = 0; t < T_DIM; ++t) {
    if ((t & (XCHUNK - 1)) == 0) {
      // barrier at end of previous v-block guarantees xbuf reads are done
      if (tid < 32) {                      // wave 0 issues one TDM DMA per chunk
        uint64_t ga = (uint64_t)(uintptr_t)x +
            (((uint64_t)t * (B_DIM * C_DIM * V_DIM)) +
             (uint64_t)b * (C_DIM * V_DIM)) * sizeof(float);
        u32x4 g0;
        g0[0] = 1u;                                            // count=1, user descriptor
        g0[1] = (uint32_t)(uintptr_t)xbuf;                     // lds_addr (bytes)
        g0[2] = (uint32_t)ga;                                  // global_addr[31:0]
        g0[3] = (uint32_t)((ga >> 32) & 0x1FFFFFFu) | (2u << 30); // addr[56:32] | type=2
        i32x8 g1;
        g1[0] = (int)(2u << 16);                               // data_size = 4B
        g1[1] = (int)((uint32_t)(C_DIM * V_DIM) << 16);        // tensor_dim0 = 512
        g1[2] = (int)((uint32_t)XCHUNK << 16);                 // tensor_dim1 = 8 rows
        g1[3] = (int)((uint32_t)(C_DIM * V_DIM) << 16);        // tile_dim0 = 512
        g1[4] = XCHUNK;                                        // tile_dim1 = 8, tile_dim2 = 0
        g1[5] = B_DIM * C_DIM * V_DIM;                         // dim0_stride = 8192 elems
        g1[6] = 0; g1[7] = 0;
        i32x4 zz; zz[0] = 0; zz[1] = 0; zz[2] = 0; zz[3] = 0;
        i32x8 z8; z8[0]=0; z8[1]=0; z8[2]=0; z8[3]=0; z8[4]=0; z8[5]=0; z8[6]=0; z8[7]=0;
        __builtin_amdgcn_tensor_load_to_lds(g0, g1, zz, zz, z8, 0);
        __builtin_amdgcn_s_wait_tensorcnt(0);
      }
      __syncthreads();
    }
    const float* xb = &xbuf[(t & (XCHUNK - 1)) * (C_DIM * V_DIM)];

    for (int vb = 0; vb < V_DIM / VT; ++vb) {
      const int v0 = vb * VT;

      // ---- phase 1: WMMA preactivation tiles (matrix pipe, fp32-exact, K 2->4 pad)
      #pragma unroll
      for (int tv = 0; tv < VT / 16; ++tv) {
        const int vcol = v0 + tv * 16 + (lane & 15);
        const float x0 = xb[vcol];             // c=0 row
        const float x1 = xb[V_DIM + vcol];     // c=1 row
        v2f bop;                               // B lanes 0-15: K=0,1 ; 16-31: K=2,3 = 0
        bop.x = (lane < 16) ? x0 : 0.f;
        bop.y = (lane < 16) ? x1 : 0.f;
        #pragma unroll
        for (int i = 0; i < 2; ++i) {
          v8f d = __builtin_amdgcn_wmma_f32_16x16x4_f32(
              false, aops[i], false, bop, (short)0, cops[i], false, false);
          const int er = (wv * 2 + i) * 16 + off;   // e row within tile
          const int vv = tv * 16 + (lane & 15);     // v column within block
          #pragma unroll
          for (int g = 0; g < 8; ++g)
            pre[(er + g) * VTP + vv] = d[g];
        }
      }
      __syncthreads();

      // ---- phase 2: serial LIF scan over v (state carried in a register)
      #pragma unroll 8
      for (int v = 0; v < VT; ++v) {
        const float inp = pre[tid * VTP + v];
        vstate = 0.5f * (vstate + inp);            // v += (inp - v)/TAU, TAU=2
        const bool fire = (vstate >= 1.0f);        // spike(v - VTH)
        sbuf[tid * VTP + v] = fire ? 1.0f : 0.0f;
        vstate = fire ? 0.0f : vstate;             // hard reset
      }
      __syncthreads();

      // ---- phase 3: coalesced float4 store of the [ETILE x VT] spike tile
      const int vq = tid & 15, rsub = tid >> 4;
      #pragma unroll
      for (int r0 = 0; r0 < ETILE; r0 += 4) {
        const int row = r0 + rsub;
        float4 val;
        val.x = sbuf[row * VTP + vq * 4 + 0];
        val.y = sbuf[row * VTP + vq * 4 + 1];
        val.z = sbuf[row * VTP + vq * 4 + 2];
        val.w = sbuf[row * VTP + vq * 4 + 3];
        size_t o = ((((size_t)t * B_DIM + b) * E_DIM) + (size_t)(e0 + row)) * V_DIM
                   + (size_t)(v0 + vq * 4);
        *reinterpret_cast<float4*>(out + o) = val;
      }
      __syncthreads();
    }
  }
}

extern "C" void kernel_launch(void* const* d_in, const int* in_sizes, int n_in,
                              void* d_out, int out_size, void* d_ws, size_t ws_size,
                              hipStream_t stream) {
  const float* x     = (const float*)d_in[0];
  const float* W     = (const float*)d_in[1];
  const float* gamma = (const float*)d_in[2];
  const float* beta  = (const float*)d_in[3];
  const float* rmean = (const float*)d_in[4];
  const float* rvar  = (const float*)d_in[5];
  float* out = (float*)d_out;

  dim3 grid(E_DIM / ETILE, B_DIM);   // 4 x 16 = 64 WGs = 128 wave32s (== #LIF chains / 32)
  dim3 block(ETILE);                 // 64 threads, 2 waves
  sps_lif_kernel<<<grid, block, 0, stream>>>(x, W, gamma, beta, rmean, rvar, out);
}